// Bridge_39505109188914
// MI455X (gfx1250) — compile-verified
//
#include <hip/hip_runtime.h>
#include <hip/hip_bf16.h>

// ---------------------------------------------------------------------------
// KNN(8) + Gaussian-softmax feature gather for MI455X (gfx1250, wave32).
//
// Score tiles computed with V_WMMA_F32_16X16X4_F32:
//   A row (per query)  = ( qx, qy, qz, 1 )
//   B col (per anchor) = ( -2px, -2py, -2pz, |p|^2 )
//   D = A*B = |p|^2 - 2 q.p   ( == d^2 - |q|^2 ; the per-query constant |q|^2
//   affects neither top-k ordering nor the softmax weights, so it is dropped )
//
// v2: software-pipelined B-fragment (removes the s_wait_dscnt 0 stall in
//     front of every WMMA) + tile-level min-tree early-out for the top-k scan.
// ---------------------------------------------------------------------------

typedef float v2f __attribute__((ext_vector_type(2)));
typedef float v4f __attribute__((ext_vector_type(4)));
typedef float v8f __attribute__((ext_vector_type(8)));

#define NANCH           16384
#define NCHUNK          8192                 // anchors resident in LDS per phase
#define NPHASE          (NANCH / NCHUNK)     // 2
#define TILES_PER_PHASE (NCHUNK / 16)        // 512
#define WAVES_PER_BLOCK 8
#define QPW             16                   // queries per wave (WMMA M)
#define FDIM            64
#define INV_TEMP        200.0f               // 1 / (2*0.05^2)

__device__ __forceinline__ void sorted_insert8(float (&s)[8], int (&id)[8],
                                               float c, int ci) {
    // replace worst if better, then one bubble pass; the pass is a no-op for
    // lanes that did not insert (their list is already sorted), so no per-lane
    // predication is needed beyond the slot-7 replacement.
    bool ins = c < s[7];
    s[7]  = ins ? c  : s[7];
    id[7] = ins ? ci : id[7];
#pragma unroll
    for (int k = 7; k >= 1; --k) {
        float a = s[k - 1], b = s[k];
        int   ia = id[k - 1], ib = id[k];
        bool  sw = b < a;
        s[k - 1]  = sw ? b  : a;
        s[k]      = sw ? a  : b;
        id[k - 1] = sw ? ib : ia;
        id[k]     = sw ? ia : ib;
    }
}

__global__ void __launch_bounds__(256)
knn_wmma_kernel(const float* __restrict__ coords,     // [B,3]
                const float* __restrict__ positions,  // [N,3]
                const float* __restrict__ features,   // [N,64]
                float* __restrict__ out)              // [B,64]
{
    __shared__ float lds_b[NCHUNK * 4];                 // 128 KB: (-2p, |p|^2)
    __shared__ float lds_scr[WAVES_PER_BLOCK * 512];    // 16 KB wave scratch

    const int tid  = threadIdx.x;
    const int wave = tid >> 5;
    const int lane = tid & 31;
    const int half = lane >> 4;     // 0: K=0,1 lanes | 1: K=2,3 lanes
    const int l16  = lane & 15;

    const int qbase = (blockIdx.x * WAVES_PER_BLOCK + wave) * QPW;

    // ---- A fragment: 16x4 f32, lanes 0-15 hold (K0,K1), lanes 16-31 (K2,K3)
    v2f afrag;
    {
        const int q = qbase + l16;
        if (half == 0) {
            afrag.x = coords[q * 3 + 0];
            afrag.y = coords[q * 3 + 1];
        } else {
            afrag.x = coords[q * 3 + 2];
            afrag.y = 1.0f;
        }
    }

    // ---- per-lane top-8 state (one query per lane after tile transpose) ----
    float s[8];
    int   id[8];
#pragma unroll
    for (int j = 0; j < 8; ++j) { s[j] = 3.0e38f; id[j] = 0; }

    float* wscr = lds_scr + wave * 512;
    int*   iscr = (int*)wscr;
    const v8f czero = {};

    for (int ph = 0; ph < NPHASE; ++ph) {
        __syncthreads();  // previous phase fully consumed
        // cooperative fill: positions -> (-2p, |p|^2) float4 in LDS
        for (int i = tid; i < NCHUNK; i += 256) {
            const int n = ph * NCHUNK + i;
            const float px = positions[n * 3 + 0];
            const float py = positions[n * 3 + 1];
            const float pz = positions[n * 3 + 2];
            v4f b;
            b.x = -2.0f * px;
            b.y = -2.0f * py;
            b.z = -2.0f * pz;
            b.w = px * px + py * py + pz * pz;
            *(v4f*)&lds_b[i * 4] = b;
        }
        __syncthreads();

        // lane's position inside a tile's B block: anchor l16, K pair
        const float* bptr = lds_b + (l16 * 4 + half * 2);

        // prologue of the software pipeline: B fragment for tile 0
        v2f bfrag = *(const v2f*)(bptr);

        for (int t = 0; t < TILES_PER_PHASE; ++t) {
            // ---- 16x16 score tile: one CDNA5 WMMA (bfrag already resident)
            v8f d = __builtin_amdgcn_wmma_f32_16x16x4_f32(
                false, afrag, false, bfrag, (short)0, czero, false, false);

            // ---- transpose tile through wave-private LDS (m-major) ----
            {
                const int mb = half * 8;  // lanes 0-15: M=0..7, 16-31: M=8..15
#pragma unroll
                for (int r = 0; r < 8; ++r)
                    wscr[(mb + r) * 16 + l16] = d[r];
            }
            asm volatile("" ::: "memory");  // DS is in-order per wave

            // lane now owns query l16, anchors [8*half, 8*half+8) of this tile
            float c[8];
            *(v4f*)&c[0] = *(const v4f*)&wscr[l16 * 16 + half * 8 + 0];
            *(v4f*)&c[4] = *(const v4f*)&wscr[l16 * 16 + half * 8 + 4];

            // ---- prefetch next tile's B fragment (in-order DS: the candidate
            //      loads above complete first, so using c waits dscnt<=1 while
            //      this load stays in flight under the VALU work below) ----
            {
                const int tn = (t + 1 < TILES_PER_PHASE) ? (t + 1) : t;
                bfrag = *(const v2f*)(bptr + tn * 64);
            }

            // ---- top-k scan with tile-level early-out ----
            const float m01 = fminf(c[0], c[1]);
            const float m23 = fminf(c[2], c[3]);
            const float m45 = fminf(c[4], c[5]);
            const float m67 = fminf(c[6], c[7]);
            const float mall = fminf(fminf(m01, m23), fminf(m45, m67));

            if (__any(mall < s[7])) {
                const int nb = ph * NCHUNK + t * 16 + half * 8;
#pragma unroll
                for (int j = 0; j < 8; ++j) {
                    const float cj = c[j];
                    if (__any(cj < s[7]))     // wave-uniform skip (rare insert)
                        sorted_insert8(s, id, cj, nb + j);
                }
            }
            asm volatile("" ::: "memory");  // loads precede next tile's stores
        }
    }

    // ---- merge the two half-wave partial lists for each query ----
    asm volatile("" ::: "memory");
#pragma unroll
    for (int j = 0; j < 8; ++j) {
        wscr[lane * 16 + j]     = s[j];
        iscr[lane * 16 + 8 + j] = id[j];
    }
    asm volatile("" ::: "memory");

    float w[8];
    if (half == 0) {
        const int p = lane + 16;  // partner lane holding same query
#pragma unroll
        for (int j = 0; j < 8; ++j)
            sorted_insert8(s, id, wscr[p * 16 + j], iscr[p * 16 + 8 + j]);

        // softmax over the 8 nearest (sorted ascending => s[0] is the min;
        // shift-invariance makes the dropped |q|^2 irrelevant)
        float tot = 0.0f;
#pragma unroll
        for (int j = 0; j < 8; ++j) {
            w[j] = __expf((s[0] - s[j]) * INV_TEMP);
            tot += w[j];
        }
        const float inv = 1.0f / tot;
#pragma unroll
        for (int j = 0; j < 8; ++j) w[j] *= inv;
    } else {
#pragma unroll
        for (int j = 0; j < 8; ++j) w[j] = 0.0f;
    }

    // ---- cooperative feature gather: whole wave per query, 2 floats/lane ----
    for (int m = 0; m < QPW; ++m) {
        float acc0 = 0.0f, acc1 = 0.0f;
#pragma unroll
        for (int k = 0; k < 8; ++k) {
            const float wk = __shfl(w[k], m, 32);
            const int   ik = __shfl(id[k], m, 32);
            const v2f fv = *(const v2f*)(features + ik * FDIM + lane * 2);
            acc0 += wk * fv.x;
            acc1 += wk * fv.y;
        }
        v2f o; o.x = acc0; o.y = acc1;
        *(v2f*)(out + (size_t)(qbase + m) * FDIM + lane * 2) = o;
    }
}

extern "C" void kernel_launch(void* const* d_in, const int* in_sizes, int n_in,
                              void* d_out, int out_size, void* d_ws, size_t ws_size,
                              hipStream_t stream) {
    (void)n_in; (void)out_size; (void)d_ws; (void)ws_size;
    const float* coords    = (const float*)d_in[0];  // [B,3]
    const float* positions = (const float*)d_in[1];  // [N,3]
    const float* features  = (const float*)d_in[2];  // [N,64]
    float* out = (float*)d_out;

    const int B = in_sizes[0] / 3;                   // 262144
    const int nblocks = B / (WAVES_PER_BLOCK * QPW); // 2048

    knn_wmma_kernel<<<nblocks, 256, 0, stream>>>(coords, positions, features, out);
}